// ResistSimulator_89163521065200
// MI455X (gfx1250) — compile-verified
//
#include <hip/hip_runtime.h>
#include <hip/hip_bf16.h>

#ifndef __has_builtin
#define __has_builtin(x) 0
#endif

namespace {
constexpr int   NZ_     = 50;
constexpr int   NSTEPS_ = 50;
constexpr float DZ_     = 20.0f;             // THICKNESS / NZ
constexpr float ZSTEP_  = 1000.0f / 49.0f;   // linspace(0,1000,50) spacing
constexpr float LOG2E_  = 1.4426950408889634f;
constexpr float C0_     = -0.0005f  * ZSTEP_ * LOG2E_;  // ALPHA0 term  -> exp2(k*C0)
constexpr float CA_     = -0.00075f * ZSTEP_ * LOG2E_;  // DILL_A term
constexpr float CB_     = -0.00005f * ZSTEP_ * LOG2E_;  // DILL_B term
constexpr float CL_     = -0.1f * LOG2E_;               // DILL_C * dt * LAMP = 0.1
constexpr float AMACK_  = 1.4264850748515f;             // 0.99^5 * 6/4
constexpr float RMIN_   = 0.8f;
constexpr float RMAX_   = 10.0f;
constexpr float DEVT_   = 15.0f;
} // namespace

__global__ __launch_bounds__(256) void resist_kernel(const float* __restrict__ img,
                                                     float* __restrict__ out,
                                                     int n) {
  const int tid = threadIdx.x;
  const int gid = blockIdx.x * 256 + tid;
  const bool active = gid < n;

  __shared__ float tile[256];

  // --- CDNA5 async global->LDS staging of this lane's pixel -----------------
#if __has_builtin(__builtin_amdgcn_global_load_async_to_lds_b32)
  if (active) {
    float* gp = const_cast<float*>(img) + gid;
    __builtin_amdgcn_global_load_async_to_lds_b32(
        (__attribute__((address_space(1))) int*)gp,
        (__attribute__((address_space(3))) int*)&tile[tid],
        /*offset=*/0, /*cpol=*/0);
  }
#if __has_builtin(__builtin_amdgcn_s_wait_asynccnt)
  __builtin_amdgcn_s_wait_asynccnt(0);
#else
  asm volatile("s_wait_asynccnt 0" ::: "memory");
#endif
#else
  if (active) tile[tid] = img[gid];
  __syncthreads();
#endif
  // Each lane consumes only its own LDS slot it requested itself, so the
  // per-wave ASYNCcnt wait is sufficient (no cross-wave barrier required).
  const float I = active ? tile[tid] : 0.0f;

  // --- exposure recurrence (loop-interchanged) + development scan -----------
  float res = 0.0f;
  float t   = DEVT_;

  for (int k = 0; k < NZ_; ++k) {
    const float kf  = (float)k;
    const float kCA = kf * CA_;
    const float kCB = kf * CB_;

    float bulk = I * __builtin_amdgcn_exp2f(kf * C0_);  // I * exp(-ALPHA0 * z_k)
    float lat  = 1.0f;
#pragma unroll
    for (int s = 0; s < NSTEPS_; ++s) {
      lat = lat * __builtin_amdgcn_exp2f(CL_ * bulk);   // lat *= exp(-0.1*bulk)
      if (s + 1 < NSTEPS_)                               // last bulk update is dead
        bulk = I * __builtin_amdgcn_exp2f(__builtin_fmaf(kCA, lat, kCB));
    }

    // Mack rate
    const float om  = 1.0f - lat;
    const float om2 = om * om;
    const float pm  = om2 * om2 * om;                    // (1-lat)^5
    float rate = pm * ((AMACK_ + 1.0f) * RMAX_) *
                 __builtin_amdgcn_rcpf(AMACK_ + pm) + RMIN_;
    rate = fminf(fmaxf(rate, RMIN_), RMAX_);

    // development step (t never goes negative: if cur>dz then t-dz/r>0)
    const float cur = rate * t;
    const bool  le  = cur <= DZ_;
    res += le ? cur : DZ_;
    t    = le ? 0.0f : t - DZ_ * __builtin_amdgcn_rcpf(rate);

    // every pixel terminates within <=8 slices (t drops by >= dz/R_MAX = 2);
    // once the whole wave is done, remaining slices are provably no-ops.
#if __has_builtin(__builtin_amdgcn_ballot_w32)
    if (__builtin_amdgcn_ballot_w32(t > 0.0f) == 0u) break;
#else
    if (__ballot(t > 0.0f) == 0ull) break;
#endif
  }

  if (active) out[gid] = res;
}

extern "C" void kernel_launch(void* const* d_in, const int* in_sizes, int n_in,
                              void* d_out, int out_size, void* d_ws, size_t ws_size,
                              hipStream_t stream) {
  const float* img = (const float*)d_in[0];
  float* out = (float*)d_out;
  const int n = in_sizes[0];          // 2*512*512 = 524288
  const int blocks = (n + 255) / 256; // exact multiple of 256
  resist_kernel<<<blocks, 256, 0, stream>>>(img, out, n);
}